// AInnoFaceLoss_29729763623289
// MI455X (gfx1250) — compile-verified
//
#include <hip/hip_runtime.h>
#include <math.h>

// ---------------------------------------------------------------------------
// AInnoFace-style detection loss, fused single pass + WMMA-based reductions.
// Stage 1 (ainno_main): per (image b, anchor c): argmax-IoU over K gt boxes
//   (gt staged in LDS as xyxy+area), focal terms for fs/ss logits, masked
//   -log(IoU) terms, per-block partial sums via V_WMMA_F32_16X16X4_F32
//   ones-reduction (exact fp32) -> 6 floats per block into d_ws.
// Stage 2 (ainno_finalize): deterministic reduction of block partials,
//   clamp counts, combine into the scalar loss.
// ---------------------------------------------------------------------------

#define MAXK 128
#define BLOCK 256
#define GX 128   // blocks per image (grid-stride over C)

typedef __attribute__((ext_vector_type(2))) float v2f;
typedef __attribute__((ext_vector_type(8))) float v8f;

// Exact fp32 sum of v across all 32 lanes of a wave using one WMMA:
// A(16x4): lanes0-15 hold {K0=v[m],K1=0}, lanes16-31 hold {K2=v[m+16],K3=0}.
// B(4x16) = ones  =>  D[m][n] = v[m] + v[m+16] for every n.
// Per lane: sum of the 8 D VGPRs = sum of 8 row-sums; xor-16 add finishes.
__device__ __forceinline__ float wave_sum32(float v) {
  v2f a; a[0] = v;    a[1] = 0.0f;
  v2f b; b[0] = 1.0f; b[1] = 1.0f;
  v8f c = {};
  v8f d = __builtin_amdgcn_wmma_f32_16x16x4_f32(
      /*neg_a=*/false, a, /*neg_b=*/false, b,
      /*c_mod=*/(short)0, c, /*reuse_a=*/false, /*reuse_b=*/false);
  float s = d[0] + d[1] + d[2] + d[3] + d[4] + d[5] + d[6] + d[7];
  s += __shfl_xor(s, 16, 32);
  return s;  // same total in all lanes
}

__device__ __forceinline__ float focal_term(float l, float t) {
  // torchvision sigmoid_focal_loss, ALPHA=0.25, GAMMA=2
  float p   = 1.0f / (1.0f + expf(-l));
  float ce  = fmaxf(l, 0.0f) - l * t + log1pf(expf(-fabsf(l)));
  float pt  = p * t + (1.0f - p) * (1.0f - t);
  float at  = 0.25f * t + 0.75f * (1.0f - t);
  float om  = 1.0f - pt;
  return at * ce * om * om;
}

__device__ __forceinline__ float iou_xyxy(float ax1, float ay1, float ax2, float ay2, float aarea,
                                          float bx1, float by1, float bx2, float by2, float barea) {
  float lx = fmaxf(ax1, bx1), ly = fmaxf(ay1, by1);
  float rx = fminf(ax2, bx2), ry = fminf(ay2, by2);
  float w = fmaxf(rx - lx, 0.0f), h = fmaxf(ry - ly, 0.0f);
  float inter = w * h;
  return inter / (aarea + barea - inter);
}

__global__ void __launch_bounds__(BLOCK)
ainno_main(const float* __restrict__ fs, const float* __restrict__ ss,
           const float* __restrict__ anchors, const float* __restrict__ gt,
           int C, int K, float* __restrict__ partials) {
  __shared__ float s_gx1[MAXK], s_gy1[MAXK], s_gx2[MAXK], s_gy2[MAXK], s_ga[MAXK];
  __shared__ float s_wave[8 * 6];

  const int tid = threadIdx.x;
  const int b   = blockIdx.y;
  const int kk  = (K < MAXK) ? K : MAXK;

  // Stage gt boxes for this image as xyxy + area (broadcast-reused C/GX times).
  for (int k = tid; k < kk; k += BLOCK) {
    const float* g = gt + ((size_t)b * K + k) * 4;
    float x = g[0], y = g[1], w = g[2], h = g[3];
    s_gx1[k] = x; s_gy1[k] = y; s_gx2[k] = x + w; s_gy2[k] = y + h;
    s_ga[k]  = w * h;
  }
  __syncthreads();

  // acc: 0=focal_ss 1=focal_fs 2=iouloss_ss 3=iouloss_fs 4=cnt_ss 5=cnt_fs
  float a0 = 0.f, a1 = 0.f, a2 = 0.f, a3 = 0.f, a4 = 0.f, a5 = 0.f;
  const size_t baseBC = (size_t)b * (size_t)C;
  const int stride = gridDim.x * BLOCK;

  for (int c = blockIdx.x * BLOCK + tid; c < C; c += stride) {
    const float* fsrow = fs + (baseBC + (size_t)c) * 6;
    const float* ssrow = ss + (baseBC + (size_t)c) * 6;
    __builtin_prefetch(fsrow, 0, 0);   // global_prefetch_b8
    __builtin_prefetch(ssrow, 0, 0);

    const float4 an = *(const float4*)(anchors + (size_t)c * 4);
    const float ax1 = an.x, ay1 = an.y, ax2 = an.x + an.z, ay2 = an.y + an.w;
    const float aarea = an.z * an.w;

    // argmax IoU over K gt boxes (first-max tie-break like jnp.argmax)
    float best_iou = -1.0f; int best_k = 0;
    for (int k = 0; k < kk; ++k) {
      float iou = iou_xyxy(ax1, ay1, ax2, ay2, aarea,
                           s_gx1[k], s_gy1[k], s_gx2[k], s_gy2[k], s_ga[k]);
      if (iou > best_iou) { best_iou = iou; best_k = k; }
    }
    const float ts = best_iou;
    const float tx1 = s_gx1[best_k], ty1 = s_gy1[best_k];
    const float tx2 = s_gx2[best_k], ty2 = s_gy2[best_k];
    const float tarea = s_ga[best_k];

    // proposal rows: 6 floats, 8B-aligned -> 3x float2
    float2 f0 = *(const float2*)(fsrow + 0);
    float2 f1 = *(const float2*)(fsrow + 2);
    float2 f2 = *(const float2*)(fsrow + 4);
    float2 s0 = *(const float2*)(ssrow + 0);
    float2 s1 = *(const float2*)(ssrow + 2);
    float2 s2 = *(const float2*)(ssrow + 4);

    a0 += focal_term(s2.x, ts);   // ss logit channel 0
    a1 += focal_term(f2.x, ts);   // fs logit channel 0

    if (ts >= 0.5f) {             // ss positive
      float iou = iou_xyxy(s0.x, s0.y, s0.x + s1.x, s0.y + s1.y, s1.x * s1.y,
                           tx1, ty1, tx2, ty2, tarea);
      a2 += -logf(iou);
      a4 += 1.0f;
    }
    if (ts >= 0.7f) {             // fs positive
      float iou = iou_xyxy(f0.x, f0.y, f0.x + f1.x, f0.y + f1.y, f1.x * f1.y,
                           tx1, ty1, tx2, ty2, tarea);
      a3 += -logf(iou);
      a5 += 1.0f;
    }
  }

  // Wave-level exact fp32 reduction via WMMA (all 256 lanes active -> EXEC ok)
  float r0 = wave_sum32(a0), r1 = wave_sum32(a1), r2 = wave_sum32(a2);
  float r3 = wave_sum32(a3), r4 = wave_sum32(a4), r5 = wave_sum32(a5);

  const int lane = tid & 31, wave = tid >> 5;
  if (lane == 0) {
    s_wave[wave * 6 + 0] = r0; s_wave[wave * 6 + 1] = r1;
    s_wave[wave * 6 + 2] = r2; s_wave[wave * 6 + 3] = r3;
    s_wave[wave * 6 + 4] = r4; s_wave[wave * 6 + 5] = r5;
  }
  __syncthreads();
  if (tid == 0) {
    float t[6] = {0, 0, 0, 0, 0, 0};
    for (int w = 0; w < BLOCK / 32; ++w)
      for (int q = 0; q < 6; ++q) t[q] += s_wave[w * 6 + q];
    float* dst = partials + ((size_t)blockIdx.y * gridDim.x + blockIdx.x) * 6;
    for (int q = 0; q < 6; ++q) dst[q] = t[q];
  }
}

__global__ void __launch_bounds__(BLOCK)
ainno_finalize(const float* __restrict__ partials, int nparts, float invN,
               float* __restrict__ out) {
  __shared__ float s_wave[8 * 6];
  float acc[6] = {0, 0, 0, 0, 0, 0};
  for (int i = threadIdx.x; i < nparts; i += BLOCK) {
    const float* p = partials + (size_t)i * 6;
    for (int q = 0; q < 6; ++q) acc[q] += p[q];
  }
  float red[6];
  for (int q = 0; q < 6; ++q) red[q] = wave_sum32(acc[q]);

  const int lane = threadIdx.x & 31, wave = threadIdx.x >> 5;
  if (lane == 0)
    for (int q = 0; q < 6; ++q) s_wave[wave * 6 + q] = red[q];
  __syncthreads();
  if (threadIdx.x == 0) {
    float t[6] = {0, 0, 0, 0, 0, 0};
    for (int w = 0; w < BLOCK / 32; ++w)
      for (int q = 0; q < 6; ++q) t[q] += s_wave[w * 6 + q];
    float ss_cnt = fmaxf(t[4], 1.0f);
    float fs_cnt = fmaxf(t[5], 1.0f);
    out[0] = t[0] * invN / ss_cnt   // ss_stc
           + t[1] * invN / fs_cnt   // fs_stc
           + t[2] / ss_cnt          // ss_str
           + t[3] / fs_cnt;         // fs_str
  }
}

extern "C" void kernel_launch(void* const* d_in, const int* in_sizes, int n_in,
                              void* d_out, int out_size, void* d_ws, size_t ws_size,
                              hipStream_t stream) {
  const float* fs      = (const float*)d_in[0];  // (B,C,6)
  const float* ss      = (const float*)d_in[1];  // (B,C,6)
  const float* anchors = (const float*)d_in[2];  // (C,4)
  const float* gt      = (const float*)d_in[3];  // (B,K,4)

  const int C = in_sizes[2] / 4;
  const int B = (int)((long long)in_sizes[0] / ((long long)C * 6));
  const int K = in_sizes[3] / (B * 4);

  float* partials = (float*)d_ws;          // GX*B blocks x 6 floats
  const int nparts = GX * B;
  const float invN = 1.0f / ((float)B * (float)C);

  ainno_main<<<dim3(GX, B), BLOCK, 0, stream>>>(fs, ss, anchors, gt, C, K, partials);
  ainno_finalize<<<1, BLOCK, 0, stream>>>(partials, nparts, invN, (float*)d_out);
}